// FeatureSampler_74062416052404
// MI455X (gfx1250) — compile-verified
//
#include <hip/hip_runtime.h>
#include <hip/hip_bf16.h>
#include <math.h>

#define B_   2
#define N_   6
#define C_   256
#define M_   900
#define EPSF 1e-5f

typedef __attribute__((ext_vector_type(2))) float v2f;
typedef __attribute__((ext_vector_type(8))) float v8f;

// Levels: (H,W) = (116,200) (58,100) (29,50) (15,25); HW = 23200, 5800, 1450, 375
// Channel-last workspace layout, per-level element offsets (bn-major blocks of HW*C):
//   off0 = 0
//   off1 = 12*23200*256 = 71270400
//   off2 = off1 + 12*5800*256 = 89088000
//   off3 = off2 + 12*1450*256 = 93542400
//   total = 94694400 elements

// ---------------------------------------------------------------------------
// Kernel 1: WMMA-batched projection.
// One wave handles one (batch, 16-point tile).
// WMMA #1: A rows 0..15 = cams 0..3 (row = cam*4+comp), K = 4 matrix columns
// WMMA #2: A rows 0..7  = cams 4..5, rows 8..15 zero
// B: 4x16, column = point, K = homogeneous coordinate.
// D[row, col] = sum_k lidar2img[cam][comp][k] * homo[point][k]
// ---------------------------------------------------------------------------
__global__ void proj_wmma_kernel(const float* __restrict__ refpts, // (B,M,3)
                                 const float* __restrict__ l2i,    // (B,N,4,4)
                                 float* __restrict__ pxw,          // (B,N,M)
                                 float* __restrict__ pyw,          // (B,N,M)
                                 float* __restrict__ vw)           // (B,N,M)
{
    const int NT   = (M_ + 15) / 16;           // 57 tiles per batch
    const int wave = blockIdx.x;               // 0 .. B_*NT-1
    const int b    = wave / NT;
    const int tile = wave % NT;

    const int lane = threadIdx.x;              // 0..31 (wave32)
    const int half = lane >> 4;                // 0 -> K={0,1}, 1 -> K={2,3}
    const int lm   = lane & 15;
    const int k0   = half * 2;
    const int k1   = k0 + 1;

    // ---- B matrix: homogeneous point for column lm ----
    const int p = tile * 16 + lm;
    float h0 = 0.f, h1 = 0.f, h2 = 0.f, h3 = 0.f;
    if (p < M_) {
        const float rx = refpts[((size_t)b * M_ + p) * 3 + 0];
        const float ry = refpts[((size_t)b * M_ + p) * 3 + 1];
        const float rz = refpts[((size_t)b * M_ + p) * 3 + 2];
        h0 = rx * 122.4f - 61.2f;   // PC_RANGE denormalization
        h1 = ry * 122.4f - 61.2f;
        h2 = rz * 20.0f  - 10.0f;
        h3 = 1.0f;
    }
    v2f bv;
    bv.x = (half == 0) ? h0 : h2;   // constant-index selects (no cndmask chain)
    bv.y = (half == 0) ? h1 : h3;

    // ---- A matrices: packed camera projection rows ----
    const int comp = lm & 3;                    // output component i
    const float* L = l2i + (size_t)b * N_ * 16; // (N,4,4) for this batch

    v2f a1;                                     // cams 0..3
    {
        const int cam = lm >> 2;
        a1.x = L[cam * 16 + comp * 4 + k0];
        a1.y = L[cam * 16 + comp * 4 + k1];
    }
    v2f a2;                                     // cams 4..5 in rows 0..7
    if (lm < 8) {
        const int cam = 4 + (lm >> 2);
        a2.x = L[cam * 16 + comp * 4 + k0];
        a2.y = L[cam * 16 + comp * 4 + k1];
    } else {
        a2.x = 0.f; a2.y = 0.f;
    }

    v8f c0 = {};
    v8f c1 = {};
    c0 = __builtin_amdgcn_wmma_f32_16x16x4_f32(false, a1, false, bv, (short)0, c0, false, false);
    c1 = __builtin_amdgcn_wmma_f32_16x16x4_f32(false, a2, false, bv, (short)0, c1, false, false);

    // ---- Extract: lane owns point p; c0 gives 2 cams, c1 gives 2 more (half==0) ----
    if (p < M_) {
        {   // c0: lanes<16 rows 0..7 (cams 0,1); lanes>=16 rows 8..15 (cams 2,3)
            const int camA = half * 2;
            float cx = c0[0], cy = c0[1], cz = c0[2];
            float zi = cz + EPSF;
            size_t idx = ((size_t)b * N_ + camA) * M_ + p;
            pxw[idx] = cx / zi; pyw[idx] = cy / zi; vw[idx] = (cz > EPSF) ? 1.f : 0.f;

            cx = c0[4]; cy = c0[5]; cz = c0[6];
            zi = cz + EPSF;
            idx = ((size_t)b * N_ + camA + 1) * M_ + p;
            pxw[idx] = cx / zi; pyw[idx] = cy / zi; vw[idx] = (cz > EPSF) ? 1.f : 0.f;
        }
        if (half == 0) {   // c1 rows 0..7: cams 4,5
            float cx = c1[0], cy = c1[1], cz = c1[2];
            float zi = cz + EPSF;
            size_t idx = ((size_t)b * N_ + 4) * M_ + p;
            pxw[idx] = cx / zi; pyw[idx] = cy / zi; vw[idx] = (cz > EPSF) ? 1.f : 0.f;

            cx = c1[4]; cy = c1[5]; cz = c1[6];
            zi = cz + EPSF;
            idx = ((size_t)b * N_ + 5) * M_ + p;
            pxw[idx] = cx / zi; pyw[idx] = cy / zi; vw[idx] = (cz > EPSF) ? 1.f : 0.f;
        }
    }
}

// ---------------------------------------------------------------------------
// Kernel 2: (B,N,C,HW) -> (B,N,HW,C) transpose via LDS tiles (coalesced both ways)
// grid: (ceil(HW/32), C/32, B*N), block (32,8)
// ---------------------------------------------------------------------------
__global__ void transpose_cl_kernel(const float* __restrict__ src,
                                    float* __restrict__ dst, int HW)
{
    __shared__ float tile[32][33];
    const int bn = blockIdx.z;
    const int c0 = blockIdx.y * 32;
    const int p0 = blockIdx.x * 32;
    const int tx = threadIdx.x, ty = threadIdx.y;

    const float* s = src + (size_t)bn * C_ * HW;
#pragma unroll
    for (int j = 0; j < 4; ++j) {
        const int c = c0 + ty + j * 8;
        const int p = p0 + tx;
        if (p < HW) tile[ty + j * 8][tx] = s[(size_t)c * HW + p];
    }
    __syncthreads();
    float* d = dst + (size_t)bn * HW * C_;
#pragma unroll
    for (int j = 0; j < 4; ++j) {
        const int p = p0 + ty + j * 8;
        const int c = c0 + tx;
        if (p < HW) d[(size_t)p * C_ + c] = tile[tx][ty + j * 8];
    }
}

// ---------------------------------------------------------------------------
// Kernel 3: bilinear sampling + level mean + masked camera mean.
// One 64-thread block (2 waves) per (b,m) query; thread = 4 channels (float4).
// Weights are uniform across the block; coordinates identical across levels
// (the gx normalization cancels inside grid_sample: x = px - 0.5).
// TR=true  -> channel-last workspace copy: each corner = global_load_b128,
//             fully coalesced (64 lanes x 16B = one 1KB contiguous burst)
// TR=false -> original channel-major layout (4 strided scalar loads / corner)
// ---------------------------------------------------------------------------
template <bool TR>
__global__ void sample_kernel(const float* __restrict__ f0, const float* __restrict__ f1,
                              const float* __restrict__ f2, const float* __restrict__ f3,
                              const float* __restrict__ tf,
                              const float* __restrict__ pxw, const float* __restrict__ pyw,
                              const float* __restrict__ vw,
                              float* __restrict__ out)
{
    const int t  = threadIdx.x;          // 0..63
    const int c4 = t * 4;                // first of 4 channels for this thread
    const int bm = blockIdx.x;           // b*M + m
    const int b  = bm / M_;
    const int m  = bm - b * M_;

    const int    Wl[4]   = {200, 100, 50, 25};
    const int    Hl[4]   = {116, 58, 29, 15};
    const int    HWl[4]  = {23200, 5800, 1450, 375};
    const size_t toff[4] = {0ull, 71270400ull, 89088000ull, 93542400ull};
    const float* fl[4]   = {f0, f1, f2, f3};

    float4 acc = make_float4(0.f, 0.f, 0.f, 0.f);
    float  wsum = 0.f;

    for (int n = 0; n < N_; ++n) {
        const size_t pidx = ((size_t)b * N_ + n) * M_ + m;
        const float v = vw[pidx];
        if (v == 0.f) continue;          // masked-out camera contributes 0
        wsum += 1.f;

        const float x   = pxw[pidx] - 0.5f;
        const float y   = pyw[pidx] - 0.5f;
        const float x0f = floorf(x);
        const float y0f = floorf(y);
        const float wx1 = x - x0f, wx0 = 1.f - wx1;
        const float wy1 = y - y0f, wy0 = 1.f - wy1;
        const float x1f = x0f + 1.f, y1f = y0f + 1.f;

#pragma unroll
        for (int l = 0; l < 4; ++l) {
            const int W = Wl[l], H = Hl[l], HW = HWl[l];
            const float Wf = (float)W, Hf = (float)H;
            const bool vx0 = (x0f >= 0.f) && (x0f < Wf);
            const bool vx1 = (x1f >= 0.f) && (x1f < Wf);
            const bool vy0 = (y0f >= 0.f) && (y0f < Hf);
            const bool vy1 = (y1f >= 0.f) && (y1f < Hf);

            const int ix0 = (int)fminf(fmaxf(x0f, 0.f), Wf - 1.f);
            const int ix1 = (int)fminf(fmaxf(x1f, 0.f), Wf - 1.f);
            const int iy0 = (int)fminf(fmaxf(y0f, 0.f), Hf - 1.f);
            const int iy1 = (int)fminf(fmaxf(y1f, 0.f), Hf - 1.f);

            if (TR) {
                const float* basef = tf + toff[l] + ((size_t)(b * N_ + n) * HW) * C_ + c4;
                auto corner = [&](int iy, int ix, float w) {
                    const float4 q = *reinterpret_cast<const float4*>(
                        basef + (size_t)(iy * W + ix) * C_);
                    acc.x += w * q.x; acc.y += w * q.y;
                    acc.z += w * q.z; acc.w += w * q.w;
                };
                if (vx0 && vy0) corner(iy0, ix0, wx0 * wy0);
                if (vx1 && vy0) corner(iy0, ix1, wx1 * wy0);
                if (vx0 && vy1) corner(iy1, ix0, wx0 * wy1);
                if (vx1 && vy1) corner(iy1, ix1, wx1 * wy1);
            } else {
                const float* basef = fl[l] + ((size_t)(b * N_ + n) * C_ + c4) * HW;
                auto corner = [&](int iy, int ix, float w) {
                    const size_t o = (size_t)(iy * W + ix);
                    acc.x += w * basef[o];
                    acc.y += w * basef[o + (size_t)HW];
                    acc.z += w * basef[o + (size_t)2 * HW];
                    acc.w += w * basef[o + (size_t)3 * HW];
                };
                if (vx0 && vy0) corner(iy0, ix0, wx0 * wy0);
                if (vx1 && vy0) corner(iy0, ix1, wx1 * wy0);
                if (vx0 && vy1) corner(iy1, ix0, wx0 * wy1);
                if (vx1 && vy1) corner(iy1, ix1, wx1 * wy1);
            }
        }
    }
    // mean over 4 levels, masked mean over cameras
    const float inv = 0.25f / (wsum + EPSF);
    float4 r;
    r.x = acc.x * inv; r.y = acc.y * inv; r.z = acc.z * inv; r.w = acc.w * inv;
    *reinterpret_cast<float4*>(out + (size_t)bm * C_ + c4) = r;
}

// ---------------------------------------------------------------------------
extern "C" void kernel_launch(void* const* d_in, const int* in_sizes, int n_in,
                              void* d_out, int out_size, void* d_ws, size_t ws_size,
                              hipStream_t stream) {
    const float* f0  = (const float*)d_in[0];
    const float* f1  = (const float*)d_in[1];
    const float* f2  = (const float*)d_in[2];
    const float* f3  = (const float*)d_in[3];
    const float* rp  = (const float*)d_in[4];
    const float* l2i = (const float*)d_in[5];
    float* out = (float*)d_out;

    float* ws = (float*)d_ws;
    const size_t BNM = (size_t)B_ * N_ * M_;      // 10800
    float* pxw = ws;
    float* pyw = ws + BNM;
    float* vw  = ws + 2 * BNM;
    float* tf  = ws + 3 * BNM;                    // 129,600 B in -> 16B aligned

    const size_t transElems = 94694400ull;        // channel-last copy of all levels
    const bool useTR = ws_size >= (3 * BNM + transElems) * sizeof(float);

    // 1) projection via WMMA: 114 waves (2 batches x 57 tiles of 16 points)
    const int NT = (M_ + 15) / 16;
    proj_wmma_kernel<<<B_ * NT, 32, 0, stream>>>(rp, l2i, pxw, pyw, vw);

    if (useTR) {
        // 2) transpose each level to channel-last
        dim3 blk(32, 8);
        transpose_cl_kernel<<<dim3((23200 + 31) / 32, C_ / 32, B_ * N_), blk, 0, stream>>>(f0, tf + 0ull,        23200);
        transpose_cl_kernel<<<dim3((5800  + 31) / 32, C_ / 32, B_ * N_), blk, 0, stream>>>(f1, tf + 71270400ull, 5800);
        transpose_cl_kernel<<<dim3((1450  + 31) / 32, C_ / 32, B_ * N_), blk, 0, stream>>>(f2, tf + 89088000ull, 1450);
        transpose_cl_kernel<<<dim3((375   + 31) / 32, C_ / 32, B_ * N_), blk, 0, stream>>>(f3, tf + 93542400ull, 375);
        // 3) coalesced sampling: float4 per lane -> global_load_b128
        sample_kernel<true><<<B_ * M_, C_ / 4, 0, stream>>>(f0, f1, f2, f3, tf, pxw, pyw, vw, out);
    } else {
        sample_kernel<false><<<B_ * M_, C_ / 4, 0, stream>>>(f0, f1, f2, f3, nullptr, pxw, pyw, vw, out);
    }
}